// STCM_54657753808944
// MI455X (gfx1250) — compile-verified
//
#include <hip/hip_runtime.h>
#include <hip/hip_bf16.h>

// ---------------------------------------------------------------------------
// CDNA5 (gfx1250) fused recurrent-net forward.
// All matmuls use V_WMMA_F32_16X16X4_F32 (wave32, full fp32 on the
// error-compounding recurrent path). Batch-parallel persistent blocks: each
// block owns 16 batch rows and runs sequential scans locally in LDS.
// Transcendentals use the CDNA5 hardware tanh when available.
// ---------------------------------------------------------------------------

#define DEVI static __device__ __forceinline__

typedef __attribute__((ext_vector_type(2))) float v2f;
typedef __attribute__((ext_vector_type(8))) float v8f;

// D = A(16x4) * B(4x16) + C, fp32. 8-arg form:
// (neg_a, A, neg_b, B, c_mod, C, reuse_a, reuse_b)
DEVI v8f wmma4(v2f a, v2f b, v8f c) {
  return __builtin_amdgcn_wmma_f32_16x16x4_f32(false, a, false, b,
                                               (short)0, c, false, false);
}

DEVI v8f zero8() {
  v8f r;
#pragma unroll
  for (int i = 0; i < 8; ++i) r[i] = 0.0f;
  return r;
}

// 8-byte (float2) load; all call sites are 8B aligned.
DEVI v2f ldf2(const float* p) { return *(const v2f*)p; }

// A-fragment: X is 16 x K row-major (LDS or global), padded stride.
// lane ln = row M, lane half hi selects K pair {4kk+2hi, 4kk+2hi+1}.
DEVI v2f afrag(const float* X, int ln, int hi, int kk, int stride) {
  return ldf2(X + ln * stride + 4 * kk + 2 * hi);
}
// B-fragment for X @ W^T, W is (Out x K) row-major: B[k][c] = W[c][k].
// Consecutive-K pair is contiguous within W's row c.
DEVI v2f bfrag(const float* W, int K, int c, int hi, int kk) {
  return ldf2(W + (size_t)c * K + 4 * kk + 2 * hi);
}

// CDNA5 has a hardware transcendental TANH (ISA ch.8 lists TANH among the
// TRANS32 ops). Use the builtin when the toolchain exposes it; otherwise a
// 4-op fast tanh via v_exp_f32. Sigmoid derived from tanh (1 trans op).
DEVI float ftanh(float x) {
#if __has_builtin(__builtin_amdgcn_tanhf)
  return __builtin_amdgcn_tanhf(x);
#else
  float e = __expf(2.0f * x);
  return 1.0f - 2.0f / (e + 1.0f);
#endif
}
DEVI float sigm(float x) { return 0.5f * ftanh(0.5f * x) + 0.5f; }

// ---------------------------------------------------------------------------
// K1: enc1 GRU, 168 steps, scalar input, H1=64. 16 batch rows / block,
// 4 waves; wave w owns hidden columns [16w,16w+16) of r/z/n.
// Recurrent weights kept as B-fragments in VGPRs for the whole loop.
// Writes the 30 needed timesteps to ht_sel[B][30][64]
// (slot 0..23 <- t=144..167 ; slot 24..29 <- t=0,24,48,72,96,120).
// ---------------------------------------------------------------------------
__global__ __launch_bounds__(128) void k1_enc1(
    const float* __restrict__ x2, const float* __restrict__ wi,
    const float* __restrict__ wh, const float* __restrict__ bi,
    const float* __restrict__ bh, float* __restrict__ ht_sel) {
  __shared__ float x2t[16 * 168];
  __shared__ float h[16 * 68];  // stride 68: conflict-free A-frag reads
  const int tid = threadIdx.x;
  const int ln = tid & 15, hi = (tid >> 4) & 1, wv = tid >> 5;
  const int b0 = blockIdx.x * 16;

  for (int i = tid; i < 16 * 168; i += 128) {
    int m = i / 168, t = i % 168;
    x2t[i] = x2[(size_t)(b0 + m) * 168 + t];
  }
  for (int i = tid; i < 16 * 68; i += 128) h[i] = 0.0f;

  const int c = wv * 16 + ln;  // hidden unit 0..63
  const float wir = wi[c], wiz = wi[64 + c], win = wi[128 + c];
  const float bir = bi[c], biz = bi[64 + c], bin = bi[128 + c];
  const float bhr = bh[c], bhz = bh[64 + c], bhn = bh[128 + c];

  v2f Br[16], Bz[16], Bn[16];
#pragma unroll
  for (int kk = 0; kk < 16; ++kk) {
    Br[kk] = bfrag(wh, 64, c, hi, kk);
    Bz[kk] = bfrag(wh, 64, 64 + c, hi, kk);
    Bn[kk] = bfrag(wh, 64, 128 + c, hi, kk);
  }
  __syncthreads();

  for (int t = 0; t < 168; ++t) {
    v8f ar = zero8(), az = zero8(), an = zero8();
#pragma unroll
    for (int kk = 0; kk < 16; ++kk) {
      v2f a = afrag(h, ln, hi, kk, 68);
      ar = wmma4(a, Br[kk], ar);
      az = wmma4(a, Bz[kk], az);
      an = wmma4(a, Bn[kk], an);
    }
    float hnew[8];
#pragma unroll
    for (int i = 0; i < 8; ++i) {
      int m = i + 8 * hi;
      float xv = x2t[m * 168 + t];
      float hold = h[m * 68 + c];
      float r = sigm(xv * wir + bir + ar[i] + bhr);
      float z = sigm(xv * wiz + biz + az[i] + bhz);
      float nn = ftanh(xv * win + bin + r * (an[i] + bhn));
      hnew[i] = (1.0f - z) * nn + z * hold;
    }
    __syncthreads();
    int slot = (t >= 144) ? (t - 144) : ((t % 24 == 0) ? (24 + t / 24) : -1);
#pragma unroll
    for (int i = 0; i < 8; ++i) {
      int m = i + 8 * hi;
      h[m * 68 + c] = hnew[i];
      if (slot >= 0)
        ht_sel[((size_t)(b0 + m) * 30 + slot) * 64 + c] = hnew[i];
    }
    __syncthreads();
  }
}

// ---------------------------------------------------------------------------
// K2: u = [x0|x1] @ ug_w ; tvg[b][n][a] = sum_l u[b][l][n] * vg_w[a][l].
// One block per batch row; u kept transposed in LDS (n-major, stride 28),
// then a (256 x 24) @ (24 x 64) WMMA GEMM. Streams the ~226 MB of inputs.
// ---------------------------------------------------------------------------
__global__ __launch_bounds__(256) void k2_tvg(
    const float* __restrict__ x0, const float* __restrict__ x1,
    const float* __restrict__ ug_w, const float* __restrict__ vg_w,
    float* __restrict__ tvg) {
  __shared__ float uT[256 * 28];
  const int tid = threadIdx.x;
  const int b = blockIdx.x;
  float ug[9];
#pragma unroll
  for (int g = 0; g < 9; ++g) ug[g] = ug_w[g];

  const int n = tid;
  for (int l = 0; l < 24; ++l) {
    const float* xp = x0 + (((size_t)b * 24 + l) * 256 + n) * 8;
    float acc = x1[((size_t)b * 24 + l) * 256 + n] * ug[8];
#pragma unroll
    for (int g = 0; g < 8; ++g) acc += xp[g] * ug[g];
    uT[n * 28 + l] = acc;
  }
  __syncthreads();

  const int ln = tid & 15, hi = (tid >> 4) & 1, wv = tid >> 5;  // 8 waves
  for (int mi = 0; mi < 2; ++mi) {
    int mt = wv + 8 * mi;  // 16 row-tiles over n
    for (int at = 0; at < 4; ++at) {
      int c = at * 16 + ln;  // a-column 0..63
      v8f acc = zero8();
#pragma unroll
      for (int kk = 0; kk < 6; ++kk) {  // K = 24
        v2f a = afrag(uT + 16 * mt * 28, ln, hi, kk, 28);
        v2f bb = bfrag(vg_w, 24, c, hi, kk);
        acc = wmma4(a, bb, acc);
      }
#pragma unroll
      for (int i = 0; i < 8; ++i) {
        int nn = 16 * mt + i + 8 * hi;
        tvg[(size_t)b * 16384 + (size_t)nn * 64 + c] = acc[i];
      }
    }
  }
}

// ---------------------------------------------------------------------------
// K3: 24-step attention-GRU scan. 16 batch rows / block, 4 waves.
// tvg tile (1 MB/step) streams from L2 (67 MB buffer << 192 MB L2).
// sh-projection and both GRU GEMMs on f32 WMMA; r/z share an accumulator
// (sigmoid(ir+hr)); n keeps input/hidden parts separate. wg_w and e2_wh
// B-fragments stay resident in VGPRs for all 24 steps.
// ---------------------------------------------------------------------------
__global__ __launch_bounds__(128) void k3_scan(
    const float* __restrict__ x1, const float* __restrict__ ht_sel,
    const float* __restrict__ tvg, const float* __restrict__ e2_wi,
    const float* __restrict__ e2_wh, const float* __restrict__ e2_bi,
    const float* __restrict__ e2_bh, const float* __restrict__ wg_w,
    const float* __restrict__ wg_b, const float* __restrict__ W_w,
    float* __restrict__ sfin, float* __restrict__ xtf) {
  __shared__ float s[16 * 68];
  __shared__ float xt[16 * 260];
  __shared__ float sh[16 * 64];
  __shared__ float htl[16 * 68];  // stride 68: WMMA A operand
  __shared__ float Wl[64];
  const int tid = threadIdx.x;
  const int ln = tid & 15, hi = (tid >> 4) & 1, wv = tid >> 5;
  const int b0 = blockIdx.x * 16;

  if (tid < 64) Wl[tid] = W_w[tid];
  for (int i = tid; i < 16 * 68; i += 128) s[i] = 0.0f;

  const int c = wv * 16 + ln;
  v2f Br[16], Bz[16], Bn[16];  // e2_wh fragments resident in VGPRs
  v2f Bs[16], Bh[16];          // wg_w fragments (s-part / h-part)
#pragma unroll
  for (int kk = 0; kk < 16; ++kk) {
    Br[kk] = bfrag(e2_wh, 64, c, hi, kk);
    Bz[kk] = bfrag(e2_wh, 64, 64 + c, hi, kk);
    Bn[kk] = bfrag(e2_wh, 64, 128 + c, hi, kk);
    Bs[kk] = bfrag(wg_w, 128, c, hi, kk);       // wg_w[c][k], k<64
    Bh[kk] = bfrag(wg_w + 64, 128, c, hi, kk);  // wg_w[c][64+k]
  }
  const float bir = e2_bi[c], biz = e2_bi[64 + c], bin = e2_bi[128 + c];
  const float bhr = e2_bh[c], bhz = e2_bh[64 + c], bhn = e2_bh[128 + c];
  const float wgbc = wg_b[c];
  __syncthreads();

  for (int t = 0; t < 24; ++t) {
    // h_t tile (= ht[:, 144+t, :]) ; prefetch next step's tile into L2/WGP$
    for (int i = tid; i < 16 * 64; i += 128) {
      int m = i >> 6, k = i & 63;
      htl[m * 68 + k] = ht_sel[((size_t)(b0 + m) * 30 + t) * 64 + k];
    }
    if (t + 1 < 24 && tid < 64) {
      const float* np =
          ht_sel + ((size_t)(b0 + (tid >> 2)) * 30 + (t + 1)) * 64 +
          (tid & 3) * 16;
      __builtin_prefetch(np, 0, 1);
    }
    __syncthreads();
    // sh = [s, h_t] @ wg_w^T + wg_b   (WMMA, K=128)
    {
      v8f sha = zero8();
#pragma unroll
      for (int kk = 0; kk < 16; ++kk)
        sha = wmma4(afrag(s, ln, hi, kk, 68), Bs[kk], sha);
#pragma unroll
      for (int kk = 0; kk < 16; ++kk)
        sha = wmma4(afrag(htl, ln, hi, kk, 68), Bh[kk], sha);
#pragma unroll
      for (int i = 0; i < 8; ++i)
        sh[(i + 8 * hi) * 64 + c] = sha[i] + wgbc;
    }
    __syncthreads();
    // a[b][n] = tanh(tvg + sh) . W_w ; x_tilde = x1_t * a
    for (int j = 0; j < 32; ++j) {
      int p = tid + 128 * j;
      int bl = p >> 8, n = p & 255;
      const float4* tp =
          (const float4*)(tvg + (((size_t)(b0 + bl)) * 256 + n) * 64);
      const float* shp = sh + bl * 64;
      float acc = 0.0f;
#pragma unroll 4
      for (int q = 0; q < 16; ++q) {
        float4 tv = tp[q];
        int ab = q * 4;
        acc += Wl[ab + 0] * ftanh(tv.x + shp[ab + 0]);
        acc += Wl[ab + 1] * ftanh(tv.y + shp[ab + 1]);
        acc += Wl[ab + 2] * ftanh(tv.z + shp[ab + 2]);
        acc += Wl[ab + 3] * ftanh(tv.w + shp[ab + 3]);
      }
      float x1v = x1[((size_t)(b0 + bl) * 24 + t) * 256 + n];
      xt[bl * 260 + n] = x1v * acc;
    }
    __syncthreads();
    // s = GRU_e2(x_tilde, s)
    v8f ar = zero8(), az = zero8(), ani = zero8(), anh = zero8();
#pragma unroll
    for (int kk = 0; kk < 16; ++kk) {  // hidden part, K=64
      v2f a = afrag(s, ln, hi, kk, 68);
      ar = wmma4(a, Br[kk], ar);
      az = wmma4(a, Bz[kk], az);
      anh = wmma4(a, Bn[kk], anh);
    }
#pragma unroll 8
    for (int kk = 0; kk < 64; ++kk) {  // input part, K=256 (weights L2-hot)
      v2f a = afrag(xt, ln, hi, kk, 260);
      ar = wmma4(a, bfrag(e2_wi, 256, c, hi, kk), ar);
      az = wmma4(a, bfrag(e2_wi, 256, 64 + c, hi, kk), az);
      ani = wmma4(a, bfrag(e2_wi, 256, 128 + c, hi, kk), ani);
    }
    float snew[8];
#pragma unroll
    for (int i = 0; i < 8; ++i) {
      int m = i + 8 * hi;
      float sold = s[m * 68 + c];
      float r = sigm(ar[i] + bir + bhr);
      float z = sigm(az[i] + biz + bhz);
      float nn = ftanh(ani[i] + bin + r * (anh[i] + bhn));
      snew[i] = (1.0f - z) * nn + z * sold;
    }
    __syncthreads();
#pragma unroll
    for (int i = 0; i < 8; ++i) s[(i + 8 * hi) * 68 + c] = snew[i];
    __syncthreads();
  }
  for (int i = tid; i < 16 * 64; i += 128) {
    int m = i >> 6, k = i & 63;
    sfin[(size_t)(b0 + m) * 64 + k] = s[m * 68 + k];
  }
  for (int i = tid; i < 16 * 256; i += 128) {
    int m = i >> 8, n = i & 255;
    xtf[(size_t)(b0 + m) * 256 + n] = xt[m * 260 + n];
  }
}

// ---------------------------------------------------------------------------
// K4: attention head + hf + decoder-GRU (WMMA) + sigmoid(fc).
// ---------------------------------------------------------------------------
DEVI int slot_of(int j) {  // ts = [161..166, 144, 120, 96, 72, 48, 24, 0]
  return (j < 6) ? (17 + j) : ((j == 6) ? 0 : (29 - (j - 7)));
}

__global__ __launch_bounds__(128) void k4_head(
    const float* __restrict__ ht_sel, const float* __restrict__ sfin,
    const float* __restrict__ xtf, const float* __restrict__ wgg_w,
    const float* __restrict__ wgg_b, const float* __restrict__ V_w,
    const float* __restrict__ V_b, const float* __restrict__ fu_w,
    const float* __restrict__ fu_b, const float* __restrict__ dec_wi,
    const float* __restrict__ dec_wh, const float* __restrict__ dec_bi,
    const float* __restrict__ dec_bh, const float* __restrict__ fc_w,
    const float* __restrict__ fc_b, float* __restrict__ out) {
  __shared__ float wl[64 * 128];  // wgg_w, later reused for fu_w
  __shared__ float sv[16 * 64];
  __shared__ float bl_[64];  // wgg_b
  __shared__ float Vl[64];
  __shared__ float ev[16 * 13];
  __shared__ float dv[16 * 64];
  __shared__ float hf[16 * 68];
  __shared__ float ov[16 * 68];
  const int tid = threadIdx.x;
  const int ln = tid & 15, hi = (tid >> 4) & 1, wv = tid >> 5;
  const int b0 = blockIdx.x * 16;

  for (int i = tid; i < 64 * 128; i += 128) wl[i] = wgg_w[i];
  if (tid < 64) { bl_[tid] = wgg_b[tid]; Vl[tid] = V_w[tid]; }
  for (int i = tid; i < 16 * 64; i += 128) {
    int m = i >> 6, k = i & 63;
    sv[i] = sfin[(size_t)(b0 + m) * 64 + k];
  }
  __syncthreads();

  // e[b][j] = V_w . tanh([hts, s] @ wgg^T + wgg_b) + V_b
  const float Vb0 = V_b[0];
  for (int p = tid; p < 208; p += 128) {
    int m = p / 13, j = p % 13;
    const float* hp = ht_sel + ((size_t)(b0 + m) * 30 + slot_of(j)) * 64;
    float hrow[64];
#pragma unroll
    for (int k = 0; k < 64; ++k) hrow[k] = hp[k];
    float acc = 0.0f;
    for (int ta = 0; ta < 64; ++ta) {
      float v = bl_[ta];
      const float* wr = wl + ta * 128;
#pragma unroll 8
      for (int k = 0; k < 64; ++k) v += hrow[k] * wr[k];
#pragma unroll 8
      for (int k = 0; k < 64; ++k) v += sv[m * 64 + k] * wr[64 + k];
      acc += Vl[ta] * ftanh(v);
    }
    ev[p] = acc + Vb0;
  }
  __syncthreads();
  // d[b][k] = sum_j hts[b][j][k] * e[b][j]; also swap wl -> fu_w
  for (int o = tid; o < 1024; o += 128) {
    int m = o >> 6, k = o & 63;
    float acc = 0.0f;
    for (int j = 0; j < 13; ++j)
      acc += ht_sel[((size_t)(b0 + m) * 30 + slot_of(j)) * 64 + k] *
             ev[m * 13 + j];
    dv[o] = acc;
  }
  for (int i = tid; i < 64 * 128; i += 128) wl[i] = fu_w[i];
  __syncthreads();
  // hf = [d, s] @ fu_w^T + fu_b
  for (int o = tid; o < 1024; o += 128) {
    int m = o >> 6, a = o & 63;
    float acc = fu_b[a];
    const float* wr = wl + a * 128;
#pragma unroll 8
    for (int k = 0; k < 64; ++k) acc += dv[m * 64 + k] * wr[k];
#pragma unroll 8
    for (int k = 0; k < 64; ++k) acc += sv[m * 64 + k] * wr[64 + k];
    hf[m * 68 + a] = acc;
  }
  __syncthreads();

  // out = GRU_dec(x_tilde_final, hf)  (one step, WMMA)
  const int c = wv * 16 + ln;
  v8f ar = zero8(), az = zero8(), ani = zero8(), anh = zero8();
#pragma unroll
  for (int kk = 0; kk < 16; ++kk) {  // hidden part, K=64
    v2f a = afrag(hf, ln, hi, kk, 68);
    ar = wmma4(a, bfrag(dec_wh, 64, c, hi, kk), ar);
    az = wmma4(a, bfrag(dec_wh, 64, 64 + c, hi, kk), az);
    anh = wmma4(a, bfrag(dec_wh, 64, 128 + c, hi, kk), anh);
  }
#pragma unroll 8
  for (int kk = 0; kk < 64; ++kk) {  // input part, K=256 (xtf from L2)
    v2f a = ldf2(xtf + (size_t)(b0 + ln) * 256 + 4 * kk + 2 * hi);
    ar = wmma4(a, bfrag(dec_wi, 256, c, hi, kk), ar);
    az = wmma4(a, bfrag(dec_wi, 256, 64 + c, hi, kk), az);
    ani = wmma4(a, bfrag(dec_wi, 256, 128 + c, hi, kk), ani);
  }
  const float bir = dec_bi[c], biz = dec_bi[64 + c], bin = dec_bi[128 + c];
  const float bhr = dec_bh[c], bhz = dec_bh[64 + c], bhn = dec_bh[128 + c];
#pragma unroll
  for (int i = 0; i < 8; ++i) {
    int m = i + 8 * hi;
    float hold = hf[m * 68 + c];
    float r = sigm(ar[i] + bir + bhr);
    float z = sigm(az[i] + biz + bhz);
    float nn = ftanh(ani[i] + bin + r * (anh[i] + bhn));
    ov[m * 68 + c] = (1.0f - z) * nn + z * hold;
  }
  __syncthreads();
  if (tid < 16) {
    float acc = fc_b[0];
#pragma unroll 8
    for (int k = 0; k < 64; ++k) acc += ov[tid * 68 + k] * fc_w[k];
    out[b0 + tid] = sigm(acc);
  }
}

// ---------------------------------------------------------------------------
extern "C" void kernel_launch(void* const* d_in, const int* in_sizes, int n_in,
                              void* d_out, int out_size, void* d_ws,
                              size_t ws_size, hipStream_t stream) {
  (void)in_sizes; (void)n_in; (void)out_size; (void)ws_size;
  const float* x0 = (const float*)d_in[0];
  const float* x1 = (const float*)d_in[1];
  const float* x2 = (const float*)d_in[2];
  const float* e1_wi = (const float*)d_in[3];
  const float* e1_wh = (const float*)d_in[4];
  const float* e1_bi = (const float*)d_in[5];
  const float* e1_bh = (const float*)d_in[6];
  const float* e2_wi = (const float*)d_in[7];
  const float* e2_wh = (const float*)d_in[8];
  const float* e2_bi = (const float*)d_in[9];
  const float* e2_bh = (const float*)d_in[10];
  const float* wg_w = (const float*)d_in[11];
  const float* wg_b = (const float*)d_in[12];
  const float* ug_w = (const float*)d_in[13];
  const float* vg_w = (const float*)d_in[14];
  const float* W_w = (const float*)d_in[15];
  const float* wgg_w = (const float*)d_in[16];
  const float* wgg_b = (const float*)d_in[17];
  const float* V_w = (const float*)d_in[18];
  const float* V_b = (const float*)d_in[19];
  const float* fu_w = (const float*)d_in[20];
  const float* fu_b = (const float*)d_in[21];
  const float* dec_wi = (const float*)d_in[22];
  const float* dec_wh = (const float*)d_in[23];
  const float* dec_bi = (const float*)d_in[24];
  const float* dec_bh = (const float*)d_in[25];
  const float* fc_w = (const float*)d_in[26];
  const float* fc_b = (const float*)d_in[27];

  float* ws = (float*)d_ws;
  float* ht_sel = ws;                        // 1024*30*64   = 1,966,080
  float* tvg = ht_sel + 1966080;             // 1024*256*64  = 16,777,216
  float* sfin = tvg + 16777216;              // 1024*64      = 65,536
  float* xtf = sfin + 65536;                 // 1024*256     = 262,144

  k1_enc1<<<64, 128, 0, stream>>>(x2, e1_wi, e1_wh, e1_bi, e1_bh, ht_sel);
  k2_tvg<<<1024, 256, 0, stream>>>(x0, x1, ug_w, vg_w, tvg);
  k3_scan<<<64, 128, 0, stream>>>(x1, ht_sel, tvg, e2_wi, e2_wh, e2_bi,
                                  e2_bh, wg_w, wg_b, W_w, sfin, xtf);
  k4_head<<<64, 128, 0, stream>>>(ht_sel, sfin, xtf, wgg_w, wgg_b, V_w, V_b,
                                  fu_w, fu_b, dec_wi, dec_wh, dec_bi, dec_bh,
                                  fc_w, fc_b, (float*)d_out);
}